// StickyHDPHMM_36301063586030
// MI455X (gfx1250) — compile-verified
//
#include <hip/hip_runtime.h>
#include <cstdint>

// Problem dimensions (fixed by the reference)
#define TT   20000
#define KK   64
#define KP1  65
#define DD   32
// Feature layout (reordered so every class is 4-aligned for WMMA K=4 steps):
//   f in [0,32)      : z_mu
//   f in [32,64)     : z_var
//   f in [64,1088)   : z_mu_i * z_mu_j   (f = 64 + i*32 + j)
//   f == 1088        : constant 1
//   f in [1089,1092) : zero pad
#define F_PAD 1092
#define NPAD  80              // 65 columns padded to 5 tiles of 16

#define LOG2PI_F 1.8378770664093453f
#define LOG2_F   0.6931471805599453f
#define NEG_BIG  (-1e30f)

typedef float v2f __attribute__((ext_vector_type(2)));
typedef float v8f __attribute__((ext_vector_type(8)));

__device__ __forceinline__ float digammaf_dev(float x) {
  // recurrence to x>=6, then asymptotic series
  float r = 0.f;
  while (x < 6.f) { r -= 1.f / x; x += 1.f; }
  float inv  = 1.f / x;
  float inv2 = inv * inv;
  return r + logf(x) - 0.5f * inv
         - inv2 * (0.0833333333333f - inv2 * (0.00833333333333f - inv2 * 0.0039682539683f));
}

__device__ __forceinline__ float logaddexp_dev(float a, float b) {
  float mx = fmaxf(a, b), mn = fminf(a, b);
  return mx + log1pf(__expf(mn - mx));
}

// ------------------------------------------------------------------
// zero a float region of workspace (W pad rows/cols must be 0)
// ------------------------------------------------------------------
__global__ void zero_kernel(float* __restrict__ p, int n) {
  int i = blockIdx.x * blockDim.x + threadIdx.x;
  if (i < n) p[i] = 0.f;
}

// ------------------------------------------------------------------
// Per-component precompute: Cholesky(Psi_k), Lam = nu*inv(Psi),
// fold everything into weight matrix W (F_PAD x NPAD, row-major):
//   rows 0..31   : (Lam mu)_i            (coeff of z_mu)
//   rows 32..63  : -0.5*Lam_ii           (coeff of z_var)
//   rows 64..1087: -0.5*Lam_ij           (coeff of z_mu_i*z_mu_j)
//   row 1088     : const_k = -0.5*(E_logdet + t3 + D*log2pi + mu'Lam mu)
// One wave32 per component k.
// ------------------------------------------------------------------
__global__ __launch_bounds__(32) void precompute_kernel(
    const float* __restrict__ mu_hat, const float* __restrict__ kappa_hat,
    const float* __restrict__ nu_hat, const float* __restrict__ Psi_hat,
    float* __restrict__ W) {
  const int k = blockIdx.x;
  const int tid = threadIdx.x;   // 0..31 == row index
  __shared__ float L[DD][DD + 1];
  __shared__ float Linv[DD][DD + 1];
  __shared__ float Lam[DD][DD + 1];
  __shared__ float smu[DD];
  __shared__ float red[DD];

  for (int j = 0; j < DD; ++j)
    L[tid][j] = Psi_hat[((size_t)k * DD + tid) * DD + j];
  smu[tid] = mu_hat[(size_t)k * DD + tid];
  __syncthreads();

  // in-place Cholesky (lower), row `tid`
  for (int j = 0; j < DD; ++j) {
    if (tid == j) {
      float s = L[j][j];
      for (int p = 0; p < j; ++p) s -= L[j][p] * L[j][p];
      L[j][j] = sqrtf(fmaxf(s, 1e-20f));
    }
    __syncthreads();
    if (tid > j) {
      float s = L[tid][j];
      for (int p = 0; p < j; ++p) s -= L[tid][p] * L[j][p];
      L[tid][j] = s / L[j][j];
    }
    __syncthreads();
  }

  // column `tid` of L^{-1} by forward substitution
  {
    float x[DD];
    for (int i = 0; i < DD; ++i) {
      if (i < tid) { x[i] = 0.f; continue; }
      float s = (i == tid) ? 1.f : 0.f;
      for (int p = tid; p < i; ++p) s -= L[i][p] * x[p];
      x[i] = s / L[i][i];
    }
    for (int i = 0; i < DD; ++i) Linv[i][tid] = x[i];
  }
  __syncthreads();

  const float nu = nu_hat[k];
  // Lam = nu * L^{-T} L^{-1}, column `tid`
  for (int i = 0; i < DD; ++i) {
    float s = 0.f;
    for (int r = 0; r < DD; ++r) s += Linv[r][i] * Linv[r][tid];
    Lam[i][tid] = nu * s;
  }
  __syncthreads();

  float v = 0.f;
  for (int jj = 0; jj < DD; ++jj) v += Lam[tid][jj] * smu[jj];
  red[tid] = v * smu[tid];

  W[(0 + tid) * NPAD + k]  = v;                     // z_mu coeff
  W[(32 + tid) * NPAD + k] = -0.5f * Lam[tid][tid]; // z_var coeff
  for (int jj = 0; jj < DD; ++jj)
    W[(64 + tid * DD + jj) * NPAD + k] = -0.5f * Lam[tid][jj]; // outer coeff
  __syncthreads();

  if (tid == 0) {
    float c = 0.f;
    for (int i = 0; i < DD; ++i) c += red[i];
    float logdet = 0.f;
    for (int i = 0; i < DD; ++i) logdet += logf(L[i][i]);
    logdet *= 2.f;
    float El = 0.f;
    for (int d = 1; d <= DD; ++d) El += digammaf_dev((nu + 1.f - (float)d) * 0.5f);
    El -= (float)DD * LOG2_F + logdet;
    float t3 = (float)DD * nu / (kappa_hat[k] * fmaxf(nu - (float)DD - 1.f, 1.f));
    W[1088 * NPAD + k] = -0.5f * (El + t3 + (float)DD * LOG2PI_F + c); // const
  }
}

// ------------------------------------------------------------------
// log_A (digamma) and log_pi. Blocks 0..63 -> rows of log_A, block 64 -> log_pi
// ------------------------------------------------------------------
__global__ __launch_bounds__(96) void trans_kernel(
    const float* __restrict__ phi_hat, const float* __restrict__ pi_star,
    float* __restrict__ logA, float* __restrict__ logpi) {
  __shared__ float s[KP1];
  __shared__ float dgsum;
  const int i = blockIdx.x;
  const int tid = threadIdx.x;
  if (i < KK) {
    if (tid < KP1) s[tid] = phi_hat[(size_t)i * KP1 + tid];
    __syncthreads();
    if (tid == 0) {
      float r = 0.f;
      for (int jj = 0; jj < KP1; ++jj) r += s[jj];
      dgsum = digammaf_dev(r);
    }
    __syncthreads();
    if (tid < KP1) logA[(size_t)i * KP1 + tid] = digammaf_dev(s[tid]) - dgsum;
  } else {
    if (tid < KP1) logpi[tid] = logf(pi_star[tid] + 1e-9f);
  }
}

// ------------------------------------------------------------------
// log_B = feat(z) @ W via V_WMMA_F32_16X16X4_F32.
// One wave per block: 16 t-rows x 80 cols (5 N-tiles), K streamed in steps of 4.
// A fragment (16x4 f32): lane l holds row M=l%16, VGPRs = K {0,1} (lanes<16)
// or K {2,3} (lanes>=16). B fragment mirrors it with N=l%16.
// Branch-free inner loops: A fragments come from aligned float2 LDS loads.
// ------------------------------------------------------------------
__device__ __forceinline__ void wmma5_step(const float* __restrict__ W, int f0,
                                           int lh, int lm, v2f a, v8f* acc) {
#pragma unroll
  for (int w = 0; w < 5; ++w) {
    v2f b;
    b.x = W[(size_t)(f0 + lh) * NPAD + w * 16 + lm];
    b.y = W[(size_t)(f0 + lh + 1) * NPAD + w * 16 + lm];
    acc[w] = __builtin_amdgcn_wmma_f32_16x16x4_f32(
        false, a, false, b, (short)0, acc[w], false, false);
  }
}

__global__ __launch_bounds__(32) void logB_wmma_kernel(
    const float* __restrict__ z_mu, const float* __restrict__ z_var,
    const float* __restrict__ W, float* __restrict__ logB) {
  __shared__ float zmu[16][DD];   // row stride 128B -> float2 loads stay 8B aligned
  __shared__ float zvar[16][DD];
  const int t0 = blockIdx.x * 16;
  const int lane = threadIdx.x;

  for (int idx = lane; idx < 16 * DD; idx += 32) {
    int r = idx >> 5, c = idx & 31;
    zmu[r][c]  = z_mu[(size_t)(t0 + r) * DD + c];
    zvar[r][c] = z_var[(size_t)(t0 + r) * DD + c];
  }
  __syncthreads();

  const int lm = lane & 15;
  const int lh = (lane < 16) ? 0 : 2;  // K-offset of this lane half
  const float* zr = zmu[lm];
  const float* vr = zvar[lm];

  v8f acc[5] = {};

  // --- segment 1: f in [0,32) : z_mu features ---
  for (int f0 = 0; f0 < 32; f0 += 4) {
    v2f a = *(const v2f*)&zr[f0 + lh];
    wmma5_step(W, f0, lh, lm, a, acc);
  }
  // --- segment 2: f in [32,64) : z_var features ---
  for (int f0 = 0; f0 < 32; f0 += 4) {
    v2f a = *(const v2f*)&vr[f0 + lh];
    wmma5_step(W, 32 + f0, lh, lm, a, acc);
  }
  // --- segment 3: f in [64,1088) : outer-product features z_i*z_j ---
  for (int i = 0; i < DD; ++i) {
    const float zi = zr[i];
    const int fbase = 64 + i * DD;
    for (int jb = 0; jb < DD; jb += 4) {
      v2f p = *(const v2f*)&zr[jb + lh];
      v2f a;
      a.x = zi * p.x;
      a.y = zi * p.y;
      wmma5_step(W, fbase + jb, lh, lm, a, acc);
    }
  }
  // --- segment 4: f0 = 1088 : constant-1 feature (rows 1089..1091 are zero) ---
  {
    v2f a;
    a.x = (lh == 0) ? 1.f : 0.f;
    a.y = 0.f;
    wmma5_step(W, 1088, lh, lm, a, acc);
  }

  const int mbase = (lane < 16) ? 0 : 8;  // D layout: VGPR r = rows M=r / M=r+8
#pragma unroll
  for (int w = 0; w < 5; ++w) {
    int ncol = w * 16 + lm;
    if (ncol < KP1) {
      for (int r = 0; r < 8; ++r) {
        int trow = t0 + mbase + r;
        if (trow < TT) logB[(size_t)trow * KP1 + ncol] = acc[w][r];
      }
    }
  }
}

// ------------------------------------------------------------------
// Forward recursion: single block, 544 threads (17 waves).
// Column j handled by aligned 8-lane group; shuffle logsumexp reductions.
// ------------------------------------------------------------------
__global__ __launch_bounds__(544) void forward_kernel(
    const float* __restrict__ logB, const float* __restrict__ logA,
    const float* __restrict__ logpi, float* __restrict__ la,
    float* __restrict__ logZ_out) {
  __shared__ float sA[KK * KP1];
  __shared__ float spi[KP1];
  __shared__ float sprev[KP1];
  const int tid = threadIdx.x;

  for (int idx = tid; idx < KK * KP1; idx += 544) sA[idx] = logA[idx];
  if (tid < KP1) spi[tid] = logpi[tid];
  __syncthreads();
  if (tid < KP1) {
    float v = spi[tid] + logB[tid];
    sprev[tid] = v;
    la[tid] = v;
  }
  __syncthreads();

  const int j = tid >> 3;
  const int sub = tid & 7;
  const bool act = (j < KP1);

  for (int t = 1; t < TT; ++t) {
    float m = NEG_BIG;
    if (act) {
      for (int i = sub; i < KP1; i += 8) {
        float v = (i < KK) ? (sprev[i] + sA[i * KP1 + j]) : (sprev[KK] + spi[j]);
        m = fmaxf(m, v);
      }
    }
    m = fmaxf(m, __shfl_xor(m, 1));
    m = fmaxf(m, __shfl_xor(m, 2));
    m = fmaxf(m, __shfl_xor(m, 4));
    float s = 0.f;
    if (act) {
      for (int i = sub; i < KP1; i += 8) {
        float v = (i < KK) ? (sprev[i] + sA[i * KP1 + j]) : (sprev[KK] + spi[j]);
        s += __expf(v - m);
      }
    }
    s += __shfl_xor(s, 1);
    s += __shfl_xor(s, 2);
    s += __shfl_xor(s, 4);
    __syncthreads();
    if (act && sub == 0) {
      float nv = logB[(size_t)t * KP1 + j] + m + __logf(s);
      sprev[j] = nv;
      la[(size_t)t * KP1 + j] = nv;
    }
    __syncthreads();
  }

  if (tid == 0) {
    float m = NEG_BIG;
    for (int i = 0; i < KP1; ++i) m = fmaxf(m, sprev[i]);
    float s = 0.f;
    for (int i = 0; i < KP1; ++i) s += __expf(sprev[i] - m);
    *logZ_out = m + __logf(s);
  }
}

// ------------------------------------------------------------------
// Backward recursion (reverse scan), same structure; row i per 8-lane group.
// ------------------------------------------------------------------
__global__ __launch_bounds__(544) void backward_kernel(
    const float* __restrict__ logB, const float* __restrict__ logA,
    const float* __restrict__ logpi, float* __restrict__ lb) {
  __shared__ float sA[KK * KP1];
  __shared__ float spi[KP1];
  __shared__ float snext[KP1];
  __shared__ float stmp[KP1];
  const int tid = threadIdx.x;

  for (int idx = tid; idx < KK * KP1; idx += 544) sA[idx] = logA[idx];
  if (tid < KP1) {
    spi[tid] = logpi[tid];
    snext[tid] = 0.f;
    lb[(size_t)(TT - 1) * KP1 + tid] = 0.f;
  }
  __syncthreads();

  const int i = tid >> 3;
  const int sub = tid & 7;
  const bool act = (i < KP1);

  for (int t = TT - 2; t >= 0; --t) {
    if (tid < KP1) stmp[tid] = logB[(size_t)(t + 1) * KP1 + tid] + snext[tid];
    __syncthreads();
    float m = NEG_BIG;
    if (act) {
      for (int jj = sub; jj < KP1; jj += 8) {
        float v = ((i < KK) ? sA[i * KP1 + jj] : spi[jj]) + stmp[jj];
        m = fmaxf(m, v);
      }
    }
    m = fmaxf(m, __shfl_xor(m, 1));
    m = fmaxf(m, __shfl_xor(m, 2));
    m = fmaxf(m, __shfl_xor(m, 4));
    float s = 0.f;
    if (act) {
      for (int jj = sub; jj < KP1; jj += 8) {
        float v = ((i < KK) ? sA[i * KP1 + jj] : spi[jj]) + stmp[jj];
        s += __expf(v - m);
      }
    }
    s += __shfl_xor(s, 1);
    s += __shfl_xor(s, 2);
    s += __shfl_xor(s, 4);
    __syncthreads();
    if (act && sub == 0) {
      float nv = m + __logf(s);
      snext[i] = nv;
      lb[(size_t)t * KP1 + i] = nv;
    }
    __syncthreads();
  }
}

// ------------------------------------------------------------------
// gamma = softmax(log_alpha + log_beta) per row; one wave per t-row.
// ------------------------------------------------------------------
__global__ __launch_bounds__(256) void gamma_kernel(
    const float* __restrict__ la, const float* __restrict__ lb,
    float* __restrict__ gamma) {
  const int wid = threadIdx.x >> 5, lane = threadIdx.x & 31;
  const int t = blockIdx.x * 8 + wid;
  if (t >= TT) return;
  const float* pa = la + (size_t)t * KP1;
  const float* pb = lb + (size_t)t * KP1;
  float v0 = pa[lane] + pb[lane];
  float v1 = pa[lane + 32] + pb[lane + 32];
  float v2 = (lane == 0) ? (pa[64] + pb[64]) : NEG_BIG;
  float m = fmaxf(fmaxf(v0, v1), v2);
  for (int o = 16; o; o >>= 1) m = fmaxf(m, __shfl_xor(m, o));
  float s = __expf(v0 - m) + __expf(v1 - m) + __expf(v2 - m);
  for (int o = 16; o; o >>= 1) s += __shfl_xor(s, o);
  float inv = 1.f / s;
  float* pg = gamma + (size_t)t * KP1;
  pg[lane]      = __expf(v0 - m) * inv;
  pg[lane + 32] = __expf(v1 - m) * inv;
  if (lane == 0) pg[64] = __expf(v2 - m) * inv;
}

// ------------------------------------------------------------------
// xi[t,i,j] = exp(temp - logsumexp(temp)),
//   temp = bb[j] + logaddexp(alpha[t,i] + logA[i,j], logpi[j]),
//   bb[j] = logB[t+1,j] + logbeta[t+1,j].  One block per t.
// ------------------------------------------------------------------
__global__ __launch_bounds__(256) void xi_kernel(
    const float* __restrict__ la, const float* __restrict__ lb,
    const float* __restrict__ logB, const float* __restrict__ logA,
    const float* __restrict__ logpi, float* __restrict__ xi) {
  __shared__ float salpha[KK];
  __shared__ float sbb[KP1];
  __shared__ float spi[KP1];
  __shared__ float stemp[KK * KP1];
  __shared__ float sredm[8];
  __shared__ float sreds[8];
  const int t = blockIdx.x;
  const int tid = threadIdx.x;

  if (tid < KK) salpha[tid] = la[(size_t)t * KP1 + tid];
  if (tid < KP1) {
    sbb[tid] = logB[(size_t)(t + 1) * KP1 + tid] + lb[(size_t)(t + 1) * KP1 + tid];
    spi[tid] = logpi[tid];
  }
  __syncthreads();

  float m = NEG_BIG;
  for (int idx = tid; idx < KK * KP1; idx += 256) {
    int i = idx / KP1;
    int jj = idx - i * KP1;
    float v = sbb[jj] + logaddexp_dev(salpha[i] + logA[idx], spi[jj]);
    stemp[idx] = v;
    m = fmaxf(m, v);
  }
  for (int o = 16; o; o >>= 1) m = fmaxf(m, __shfl_xor(m, o));
  if ((tid & 31) == 0) sredm[tid >> 5] = m;
  __syncthreads();
  if (tid == 0) {
    float mm = sredm[0];
    for (int w = 1; w < 8; ++w) mm = fmaxf(mm, sredm[w]);
    sredm[0] = mm;
  }
  __syncthreads();
  m = sredm[0];

  float s = 0.f;
  for (int idx = tid; idx < KK * KP1; idx += 256) s += __expf(stemp[idx] - m);
  for (int o = 16; o; o >>= 1) s += __shfl_xor(s, o);
  if ((tid & 31) == 0) sreds[tid >> 5] = s;
  __syncthreads();
  if (tid == 0) {
    float ss = 0.f;
    for (int w = 0; w < 8; ++w) ss += sreds[w];
    sreds[0] = ss;
  }
  __syncthreads();
  const float norm = m + __logf(sreds[0]);

  float* po = xi + (size_t)t * KK * KP1;
  for (int idx = tid; idx < KK * KP1; idx += 256)
    po[idx] = __expf(stemp[idx] - norm);
}

// ------------------------------------------------------------------
extern "C" void kernel_launch(void* const* d_in, const int* in_sizes, int n_in,
                              void* d_out, int out_size, void* d_ws, size_t ws_size,
                              hipStream_t stream) {
  const float* z_mu   = (const float*)d_in[0];
  const float* z_var  = (const float*)d_in[1];
  const float* mu_hat = (const float*)d_in[2];
  const float* kappa  = (const float*)d_in[3];
  const float* nu     = (const float*)d_in[4];
  const float* Psi    = (const float*)d_in[5];
  const float* phi    = (const float*)d_in[6];
  const float* pis    = (const float*)d_in[7];

  float* out   = (float*)d_out;
  float* gamma = out;                                 // T*Kp1
  float* xi    = out + (size_t)TT * KP1;              // (T-1)*K*Kp1
  float* logZ  = xi + (size_t)(TT - 1) * KK * KP1;    // 1

  float* ws    = (float*)d_ws;
  float* W     = ws;                                  // F_PAD*NPAD
  float* logA  = W + (size_t)F_PAD * NPAD;            // KK*KP1
  float* logpi = logA + (size_t)KK * KP1;             // KP1
  float* logB  = logpi + KP1;                         // TT*KP1
  float* la    = logB + (size_t)TT * KP1;             // TT*KP1
  float* lb    = la + (size_t)TT * KP1;               // TT*KP1

  const int wn = F_PAD * NPAD;
  zero_kernel<<<(wn + 255) / 256, 256, 0, stream>>>(W, wn);
  precompute_kernel<<<KP1, 32, 0, stream>>>(mu_hat, kappa, nu, Psi, W);
  trans_kernel<<<KP1, 96, 0, stream>>>(phi, pis, logA, logpi);
  logB_wmma_kernel<<<TT / 16, 32, 0, stream>>>(z_mu, z_var, W, logB);
  forward_kernel<<<1, 544, 0, stream>>>(logB, logA, logpi, la, logZ);
  backward_kernel<<<1, 544, 0, stream>>>(logB, logA, logpi, lb);
  gamma_kernel<<<(TT + 7) / 8, 256, 0, stream>>>(la, lb, gamma);
  xi_kernel<<<TT - 1, 256, 0, stream>>>(la, lb, logB, logA, logpi, xi);
}